// VectorQuantizerEMA_72215580115688
// MI455X (gfx1250) — compile-verified
//
#include <hip/hip_runtime.h>
#include <math.h>

typedef __attribute__((ext_vector_type(2))) float f2;
typedef __attribute__((ext_vector_type(4))) float f4;
typedef __attribute__((ext_vector_type(2))) float v2f;
typedef __attribute__((ext_vector_type(8))) float v8f;

typedef __attribute__((ext_vector_type(4))) unsigned int u32x4;
typedef __attribute__((ext_vector_type(8))) int i32x8;
typedef __attribute__((ext_vector_type(4))) int i32x4;

// problem sizes
#define NUM_EMB 512
#define EMB_DIM 64
#define NPTS    131072      // 32*64*64 points
#define X_ELEMS 8388608     // 32*64*64*64

// LDS embedding row stride (64 data + 4 pad floats -> conflict-free b64 reads)
#define EROW 68

// d_out layout (flat f32, tuple return order)
#define OFF_LOSS  0L
#define OFF_QOUT  1L
#define OFF_PERP  8388609L
#define OFF_ENC   8388610L
#define OFF_QFLAT 75497474L

// d_ws layout (bytes)
#define WS_LOSS   0
#define WS_COUNTS 256
#define WS_ENORM  4096
#define WS_IDX    8192

// ---------------------------------------------------------------------------
// 1) init: zero accumulators, precompute ||e_k||^2
__global__ void vq_init(const float* __restrict__ E, float* lossAcc,
                        int* counts, float* enorm) {
  int k = threadIdx.x;                 // 512 threads
  if (k == 0) *lossAcc = 0.0f;
  const f4* row = (const f4*)(E + k * EMB_DIM);
  float s = 0.0f;
#pragma unroll
  for (int i = 0; i < 16; ++i) {
    f4 v = row[i];
    s += v.x * v.x + v.y * v.y + v.z * v.z + v.w * v.w;
  }
  enorm[k] = s;
  counts[k] = 0;
}

// ---------------------------------------------------------------------------
// 2) zero the one-hot encodings region (268 MB) with NT b64 stores
__global__ void vq_zero_enc(f2* __restrict__ enc2) {
  const long total = 33554432L;        // 67,108,864 floats / 2
  long i = (long)blockIdx.x * blockDim.x + threadIdx.x;
  const long stride = (long)gridDim.x * blockDim.x;
  f2 z; z.x = 0.0f; z.y = 0.0f;
  for (; i < total; i += stride) __builtin_nontemporal_store(z, &enc2[i]);
}

// ---------------------------------------------------------------------------
// 3) main: TDM-stage embedding to LDS once per block, then per 16-point tile
//    run the f32 WMMA distance GEMM against all 512 codes from LDS.
//    D[code][pt] tile = A(E tile 16x64) x B(X^T tile 64x16), K-chained 16x16x4
__global__ void __launch_bounds__(256) vq_main(
    const float* __restrict__ X,      // NCHW input (points = w-major!)
    const float* __restrict__ E,
    const float* __restrict__ enorm,
    int* __restrict__ idxOut,
    float* __restrict__ lossAcc,
    int* __restrict__ counts) {
  __shared__ float sE[NUM_EMB * EROW];   // 512 x 68 floats = 139,264 B (< 320 KB)

  // --- Tensor Data Mover: load embedding (512x64 f32, row-padded to 68) ---
  if (threadIdx.x < 32) {                 // wave 0 only issues the TDM op
    unsigned lds0 = (unsigned)(unsigned long long)(void*)&sE[0];
    unsigned long long ga = (unsigned long long)(const void*)E;

    u32x4 g0;
    g0[0] = 1u;                                            // count=1, user D#
    g0[1] = lds0;                                          // lds_addr (bytes)
    g0[2] = (unsigned)(ga & 0xFFFFFFFFu);                  // global_addr[31:0]
    g0[3] = (unsigned)((ga >> 32) & 0x1FFFFFFu)            // global_addr[56:32]
          | (2u << 30);                                    // type = 2 ("image")

    i32x8 g1;
    g1[0] = (2 << 16)      // data_size = 4 bytes
          | (1 << 20)      // pad_enable
          | (5 << 22)      // pad_interval: every 64 DWORDs
          | (3 << 25);     // pad_amount: 4 DWORDs -> row stride 68 floats
    g1[1] = (EMB_DIM << 16);          // tensor_dim0 = 64 (bits 79:48, low half)
    g1[2] = (NUM_EMB << 16);          // tensor_dim0 hi = 0 ; tensor_dim1 = 512
    g1[3] = (EMB_DIM << 16);          // tensor_dim1 hi = 0 ; tile_dim0 = 64
    g1[4] = NUM_EMB;                  // tile_dim1 = 512 ; tile_dim2 = 0
    g1[5] = EMB_DIM;                  // tensor_dim0_stride = 64 (low 32)
    g1[6] = 0;                        // stride0 hi ; tensor_dim1_stride lo
    g1[7] = 0;
    i32x4 gz4 = {0, 0, 0, 0};         // 2D tensor: groups 2/3 unused
    i32x8 gz8 = {0, 0, 0, 0, 0, 0, 0, 0};

    __builtin_amdgcn_tensor_load_to_lds(g0, g1, gz4, gz4, gz8, 0);
    __builtin_amdgcn_s_wait_tensorcnt(0);
  }

  const int lane = threadIdx.x & 31;
  const int m    = lane & 15;          // point-in-tile (B column) / code row for A
  const int hf   = lane >> 4;          // half-wave: selects K pair {0,1} vs {2,3}
  const int tile = blockIdx.x * 8 + (threadIdx.x >> 5);
  const int p    = tile * 16 + m;      // this lane's point
  const int n    = p >> 12;
  const int xoff = n * 262144 + (p & 4095);  // element offset of (p, d=0) in NCHW

  // --- load B (X^T tile): lane holds dims {kb*4+hf*2, +1} of its point.
  //     For a fixed dim, 16 consecutive points are contiguous in NCHW memory.
  float b[32];
  float xsq = 0.0f;
#pragma unroll
  for (int kb = 0; kb < 16; ++kb) {
    const int d0 = kb * 4 + hf * 2;
    float b0 = X[xoff + d0 * 4096];
    float b1 = X[xoff + d0 * 4096 + 4096];
    b[2 * kb]     = b0;
    b[2 * kb + 1] = b1;
    xsq = __builtin_fmaf(b0, b0, xsq);
    xsq = __builtin_fmaf(b1, b1, xsq);
  }

  __syncthreads();                     // embedding resident in LDS for all waves

  float bestVal = 3.4e38f;
  int   bestIdx = 0;

  for (int ct = 0; ct < 32; ++ct) {    // 32 code tiles of 16
    const int cbase = ct << 4;
    // A tile from LDS: lane's embedding row, conflict-free ds_load_b64 reads
    const f2* A2  = (const f2*)(sE + (cbase + m) * EROW);
    const f4* en4 = (const f4*)(enorm + cbase + hf * 8);
    f4 e0 = en4[0], e1 = en4[1];

    v8f acc = {0.f, 0.f, 0.f, 0.f, 0.f, 0.f, 0.f, 0.f};
#pragma unroll
    for (int kb = 0; kb < 16; ++kb) {
      f2 av = A2[kb * 2 + hf];         // dims {kb*4+hf*2, +1} of code cbase+m
      v2f a;  a[0] = av.x;       a[1] = av.y;
      v2f bv; bv[0] = b[2 * kb]; bv[1] = b[2 * kb + 1];
      acc = __builtin_amdgcn_wmma_f32_16x16x4_f32(
          /*neg_a=*/false, a, /*neg_b=*/false, bv,
          /*c_mod=*/(short)0, acc, /*reuse_a=*/false, /*reuse_b=*/false);
    }

    // C/D layout: VGPR r holds (code = cbase + hf*8 + r, point = m)
    float en[8] = {e0.x, e0.y, e0.z, e0.w, e1.x, e1.y, e1.z, e1.w};
#pragma unroll
    for (int r = 0; r < 8; ++r) {
      float s = __builtin_fmaf(-2.0f, acc[r], en[r]);  // ||e||^2 - 2 x.e
      int code = cbase + hf * 8 + r;
      if (s < bestVal) { bestVal = s; bestIdx = code; }  // strict < => first index wins
    }
  }

  // combine the two half-wave candidates for the same point (tie -> lower index)
  float oVal = __shfl_xor(bestVal, 16, 32);
  int   oIdx = __shfl_xor(bestIdx, 16, 32);
  if (oVal < bestVal || (oVal == bestVal && oIdx < bestIdx)) {
    bestVal = oVal; bestIdx = oIdx;
  }
  float xsqT = xsq + __shfl_xor(xsq, 16, 32);   // full ||x||^2 of this point
  float dist = xsqT + bestVal;                  // exact min squared distance

  if (hf == 0) {
    idxOut[p] = bestIdx;
    atomicAdd(&counts[bestIdx], 1);
  }
  // sum dist over the 16 points of this tile (lanes 0..15), one atomic per wave
  float s = dist;
  s += __shfl_xor(s, 8, 32);
  s += __shfl_xor(s, 4, 32);
  s += __shfl_xor(s, 2, 32);
  s += __shfl_xor(s, 1, 32);
  if (lane == 0) atomicAdd(lossAcc, s);
}

// ---------------------------------------------------------------------------
// 4) quantized_flat = E[idx] (coalesced) + scatter one-hot 1.0s
__global__ void vq_flat(const float* __restrict__ E, const int* __restrict__ idx,
                        float* __restrict__ out) {
  const int lane = threadIdx.x & 31;
  const int p = blockIdx.x * 8 + (threadIdx.x >> 5);  // one wave per point
  const int id = idx[p];
  const f2* E2 = (const f2*)E;
  f2 v = E2[id * 32 + lane];
  f2* Q2 = (f2*)(out + OFF_QFLAT);
  __builtin_nontemporal_store(v, &Q2[(long)p * 32 + lane]);
  if (lane == 0) out[OFF_ENC + (long)p * 512 + id] = 1.0f;
}

// ---------------------------------------------------------------------------
// 5) quantized back to NCHW (straight-through forward value == quantized)
__global__ void vq_nchw(const float* __restrict__ E, const int* __restrict__ idx,
                        float* __restrict__ qout) {
  const int t = blockIdx.x * blockDim.x + threadIdx.x;  // ((n*64+d)*64+h)*64+w
  const int w = t & 63, h = (t >> 6) & 63, d = (t >> 12) & 63, n = t >> 18;
  const int p = n * 4096 + h * 64 + w;
  const int id = idx[p];
  float v = E[id * EMB_DIM + d];
  __builtin_nontemporal_store(v, &qout[t]);
}

// ---------------------------------------------------------------------------
// 6) scalars: loss and perplexity
__global__ void vq_final(const float* lossAcc, const int* counts,
                         float* __restrict__ out) {
  __shared__ float red[512];
  int k = threadIdx.x;
  float pk = (float)counts[k] * (1.0f / 131072.0f);
  red[k] = pk * logf(pk + 1e-10f);
  __syncthreads();
  for (int s = 256; s > 0; s >>= 1) {
    if (k < s) red[k] += red[k + s];
    __syncthreads();
  }
  if (k == 0) {
    out[OFF_PERP] = expf(-red[0]);
    out[OFF_LOSS] = 0.25f * (*lossAcc) * (1.0f / (float)X_ELEMS);
  }
}

// ---------------------------------------------------------------------------
extern "C" void kernel_launch(void* const* d_in, const int* in_sizes, int n_in,
                              void* d_out, int out_size, void* d_ws, size_t ws_size,
                              hipStream_t stream) {
  const float* X = (const float*)d_in[0];   // (32,64,64,64) f32 NCHW
  const float* E = (const float*)d_in[1];   // (512,64) f32
  float* out = (float*)d_out;

  char* ws = (char*)d_ws;
  float* lossAcc = (float*)(ws + WS_LOSS);
  int*   counts  = (int*)(ws + WS_COUNTS);
  float* enorm   = (float*)(ws + WS_ENORM);
  int*   idx     = (int*)(ws + WS_IDX);     // 131072 ints

  vq_init<<<1, 512, 0, stream>>>(E, lossAcc, counts, enorm);
  vq_zero_enc<<<8192, 256, 0, stream>>>((f2*)(out + OFF_ENC));
  vq_main<<<1024, 256, 0, stream>>>(X, E, enorm, idx, lossAcc, counts);
  vq_flat<<<16384, 256, 0, stream>>>(E, idx, out);
  vq_nchw<<<32768, 256, 0, stream>>>(E, idx, out + OFF_QOUT);
  vq_final<<<1, 512, 0, stream>>>(lossAcc, counts, out);
}